// AMNTDDA_82025285419302
// MI455X (gfx1250) — compile-verified
//
#include <hip/hip_runtime.h>
#include <math.h>

#define DIM 128
#define N_NODE 16384
#define N_PAIR 262144
#define PAIR_DIM 772
#define HID_M 384
#define CPT 256

typedef __attribute__((ext_vector_type(2))) float v2f;
typedef __attribute__((ext_vector_type(8))) float v8f;

__device__ __forceinline__ float gelu_f(float x) {
  return 0.5f * x * (1.0f + erff(x * 0.70710678118654752440f));
}

struct FuseP { const float *b1,*b2,*ln_b,*ln_g,*out_b,*out_g,*w1,*w2; };
struct GateP { const float *nb1,*nb2,*nln_b,*nln_g,*nw1,*nw2,*out_b,*out_g,
               *pb1,*pb2,*pln_b,*pln_g,*pw1,*pw2; };
struct PairP { const float *b1,*b2,*b3,*bb,*bw,*ln_b,*ln_g,*sb,*scale,*sw,
               *w1,*w2,*w3,*tb1,*tb2,*topo_scale,*tw1,*tw2; };

// ---------------- block-wide sum over 128 threads ----------------
__device__ __forceinline__ float bsum128(float v, float* r, int t) {
  r[t] = v; __syncthreads();
  #pragma unroll
  for (int o = 64; o > 0; o >>= 1) {
    if (t < o) r[t] += r[t + o];
    __syncthreads();
  }
  float s = r[0]; __syncthreads();
  return s;
}

// ================= node pipeline: fusion + gate (VALU; <4% of FLOPs) ========
__global__ __launch_bounds__(128) void node_kernel(
    const float* __restrict__ views,   // [N][3][128]
    const float* __restrict__ hgt,     // [N][128]
    FuseP F, GateP G,
    float* __restrict__ out)           // [N][128]
{
  __shared__ float sV[3][128];
  __shared__ float sX[384];
  __shared__ float sH[128];
  __shared__ float sR[128];
  const int n = blockIdx.x, t = threadIdx.x;

  sV[0][t] = views[n*384 + t];
  sV[1][t] = views[n*384 + 128 + t];
  sV[2][t] = views[n*384 + 256 + t];
  __syncthreads();

  // ---- fusion: per-view score ----
  float sc[3];
  for (int v = 0; v < 3; v++) {
    float x   = sV[v][t];
    float m   = bsum128(x, sR, t) * (1.f/128.f);
    float d   = x - m;
    float var = bsum128(d*d, sR, t) * (1.f/128.f);
    float xn  = d * rsqrtf(var + 1e-5f) * F.ln_g[t] + F.ln_b[t];
    sX[t] = xn; __syncthreads();
    float part = 0.f;
    if (t < 64) {
      float acc = F.b1[t];
      for (int k = 0; k < 128; k++) acc += sX[k] * F.w1[k*64 + t];
      part = gelu_f(acc) * F.w2[t];
    }
    sc[v] = bsum128(part, sR, t) + F.b2[0];
  }
  float mx = fmaxf(sc[0], fmaxf(sc[1], sc[2]));
  float e0 = expf(sc[0]-mx), e1 = expf(sc[1]-mx), e2 = expf(sc[2]-mx);
  float wi = 1.f / (e0 + e1 + e2);
  float fu = (e0*sV[0][t] + e1*sV[1][t] + e2*sV[2][t]) * wi;
  float m  = bsum128(fu, sR, t) * (1.f/128.f);
  float d  = fu - m;
  float var = bsum128(d*d, sR, t) * (1.f/128.f);
  float sim = d * rsqrtf(var + 1e-5f) * F.out_g[t] + F.out_b[t];
  float h   = hgt[n*128 + t];

  // ---- gate: topo path (LN over 256 -> pw1 -> gelu -> pw2) ----
  float mu  = bsum128(sim + h, sR, t) * (1.f/256.f);
  float d0 = sim - mu, d1 = h - mu;
  float vv = bsum128(d0*d0 + d1*d1, sR, t) * (1.f/256.f);
  float iv = rsqrtf(vv + 1e-5f);
  sX[t]     = d0*iv*G.pln_g[t]     + G.pln_b[t];
  sX[128+t] = d1*iv*G.pln_g[128+t] + G.pln_b[128+t];
  __syncthreads();
  float acc = G.pb1[t];
  for (int k = 0; k < 256; k++) acc += sX[k] * G.pw1[k*128 + t];
  __syncthreads();
  sH[t] = gelu_f(acc); __syncthreads();
  float topo = G.pb2[t];
  for (int k = 0; k < 128; k++) topo += sH[k] * G.pw2[k*128 + t];

  // ---- gate: n path (LN over 384 -> nw1 -> gelu -> nw2 -> sigmoid) ----
  float mu3 = bsum128(sim + h + topo, sR, t) * (1.f/384.f);
  float z0 = sim - mu3, z1 = h - mu3, z2 = topo - mu3;
  float v3 = bsum128(z0*z0 + z1*z1 + z2*z2, sR, t) * (1.f/384.f);
  float iv3 = rsqrtf(v3 + 1e-5f);
  sX[t]     = z0*iv3*G.nln_g[t]     + G.nln_b[t];
  sX[128+t] = z1*iv3*G.nln_g[128+t] + G.nln_b[128+t];
  sX[256+t] = z2*iv3*G.nln_g[256+t] + G.nln_b[256+t];
  __syncthreads();
  acc = G.nb1[t];
  for (int k = 0; k < 384; k++) acc += sX[k] * G.nw1[k*128 + t];
  __syncthreads();
  sH[t] = gelu_f(acc); __syncthreads();
  acc = G.nb2[t];
  for (int k = 0; k < 128; k++) acc += sH[k] * G.nw2[k*128 + t];
  float gt = 1.f / (1.f + expf(-acc));
  float fz = sim + gt * (h + topo - sim);
  float m4 = bsum128(fz, sR, t) * (1.f/128.f);
  float d4 = fz - m4;
  float v4 = bsum128(d4*d4, sR, t) * (1.f/128.f);
  out[n*128 + t] = d4 * rsqrtf(v4 + 1e-5f) * G.out_g[t] + G.out_b[t];
}

// ================= pair pipeline: WMMA fp32 GEMMs ==========================
// LDS strides padded for bank-conflict-free A-fragment reads (stride%64 in {4,12}).
#define LD_PDPI 132     // pd/pi rows
#define LD_A    780     // full/LN(full) rows (772 + 8)
#define LD_H1   388     // h1 rows (384 + 4)
#define LD_H2   260     // h2 rows (256 + 4)
#define OFF_PD  0
#define OFF_PI  (32*LD_PDPI)                 // 4224
#define OFF_A   (OFF_PI + 32*LD_PDPI)        // 8448
#define OFF_H1  (OFF_A + 32*LD_A)            // 33408 (also TMP[2][32][132])
#define OFF_H2  (OFF_H1 + 32*LD_H1)          // 45824 (also Ht[32][132])
#define OFF_RED (OFF_H2 + 32*LD_H2)          // 54144
#define OFF_SC  (OFF_RED + 256)              // 54400
#define SMEM_FLOATS (OFF_SC + 128)           // 54528
#define SMEM_BYTES  (SMEM_FLOATS * 4)        // 218112 B (<= 320KB/WGP)

__device__ __forceinline__ v8f wmma4(v2f a, v2f b, v8f c) {
  // V_WMMA_F32_16X16X4_F32: fp32 A/B, fp32 accumulate (matches reference math)
  return __builtin_amdgcn_wmma_f32_16x16x4_f32(false, a, false, b, (short)0, c,
                                               false, false);
}

// One 16-col tile, two 16-row M-halves sharing each B fragment (B row-major [K][ldb])
__device__ __forceinline__ void gemm2m(v8f& c0, v8f& c1,
    const float* aLds, int lda, const float* __restrict__ B, int ldb,
    int n0, int K, int half, int l16) {
  const float* a0 = aLds + l16*lda + half*2;
  const float* a1 = aLds + (l16+16)*lda + half*2;
  const float* bp = B + n0 + l16 + (half*2)*ldb;
  #pragma unroll 2
  for (int k = 0; k < K; k += 4) {
    v2f b; b.x = bp[0]; b.y = bp[ldb]; bp += 4*ldb;
    v2f a; a.x = a0[k]; a.y = a0[k+1];
    v2f a2; a2.x = a1[k]; a2.y = a1[k+1];
    c0 = wmma4(a, b, c0);
    c1 = wmma4(a2, b, c1);
  }
}

// Same, but B given transposed: Bt[N][K] (used for pi @ bw[k]^T)
__device__ __forceinline__ void gemm2m_bt(v8f& c0, v8f& c1,
    const float* aLds, int lda, const float* __restrict__ Bt, int ldbt,
    int n0, int K, int half, int l16) {
  const float* a0 = aLds + l16*lda + half*2;
  const float* a1 = aLds + (l16+16)*lda + half*2;
  const float* bp = Bt + (n0+l16)*ldbt + half*2;
  #pragma unroll 2
  for (int k = 0; k < K; k += 4) {
    v2f b; b.x = bp[k]; b.y = bp[k+1];
    v2f a; a.x = a0[k]; a.y = a0[k+1];
    v2f a2; a2.x = a1[k]; a2.y = a1[k+1];
    c0 = wmma4(a, b, c0);
    c1 = wmma4(a2, b, c1);
  }
}

// C/D layout: VGPR r -> row r (lanes 0-15) / row r+8 (lanes 16-31), col = n0+l16
__device__ __forceinline__ void store2m(float* dst, int ldd, int n0,
    v8f c0, v8f c1, const float* bias, int half, int l16, int act) {
  float bv = bias ? bias[n0 + l16] : 0.f;
  #pragma unroll
  for (int r = 0; r < 8; r++) {
    int mrow = r + half*8;
    float v0 = c0[r] + bv; if (act) v0 = gelu_f(v0);
    float v1 = c1[r] + bv; if (act) v1 = gelu_f(v1);
    dst[mrow*ldd + n0 + l16]      = v0;
    dst[(mrow+16)*ldd + n0 + l16] = v1;
  }
}

// 8-lane per-pair reduction via LDS (pp = tid>>3, ll = tid&7)
__device__ __forceinline__ float pred8(float v, float* red, int pp, int ll) {
  red[pp*8 + ll] = v; __syncthreads();
  if (ll < 4) red[pp*8+ll] += red[pp*8+ll+4]; __syncthreads();
  if (ll < 2) red[pp*8+ll] += red[pp*8+ll+2]; __syncthreads();
  if (ll < 1) red[pp*8+ll] += red[pp*8+ll+1]; __syncthreads();
  float r = red[pp*8]; __syncthreads();
  return r;
}

__global__ __launch_bounds__(256, 1) void pair_kernel(
    const float* __restrict__ drugE, const float* __restrict__ disE,
    const int* __restrict__ sample, PairP P, float* __restrict__ out)
{
  extern __shared__ float smem[];
  float* PD  = smem + OFF_PD;    // [32][132]
  float* PI  = smem + OFF_PI;    // [32][132]
  float* A   = smem + OFF_A;     // [32][780]  full -> LN(full)
  float* H1  = smem + OFF_H1;    // [32][388]  (later TMP[2][32][132])
  float* H2  = smem + OFF_H2;    // [32][260]  (first Ht[32][132])
  float* RED = smem + OFF_RED;   // [32][8]
  float* SC  = smem + OFF_SC;    // [32][4]

  const int tid  = threadIdx.x;
  const int pp   = tid >> 3, ll = tid & 7;          // 32 pairs x 8 lanes
  const int lane = tid & 31, wv = tid >> 5;          // 8 waves
  const int half = lane >> 4, l16 = lane & 15;
  const int pg   = blockIdx.x * 32 + pp;

  // ---- gather pd/pi (float4, 4 per lane) ----
  {
    int di = sample[2*pg], si = sample[2*pg + 1];
    const float4* s0 = (const float4*)(drugE + di*128);
    const float4* s1 = (const float4*)(disE  + si*128);
    float4* t0 = (float4*)(PD + pp*LD_PDPI);
    float4* t1 = (float4*)(PI + pp*LD_PDPI);
    #pragma unroll
    for (int q = 0; q < 4; q++) { t0[ll + 8*q] = s0[ll + 8*q]; t1[ll + 8*q] = s1[ll + 8*q]; }
  }
  __syncthreads();

  // ---- topo hidden: Ht = cat(pd,pi) @ tw1  (WMMA, K=256) ----
  {
    float* Ht = H2;
    int n0 = wv * 16;
    v8f c0 = {0,0,0,0,0,0,0,0}, c1 = {0,0,0,0,0,0,0,0};
    gemm2m(c0, c1, PD, LD_PDPI, P.tw1,           128, n0, 128, half, l16);
    gemm2m(c0, c1, PI, LD_PDPI, P.tw1 + 128*128, 128, n0, 128, half, l16);
    store2m(Ht, LD_PDPI, n0, c0, c1, nullptr, half, l16, 0);
  }
  __syncthreads();

  // ---- per-pair scalars ----
  float dotp = 0, nx2 = 0, ny2 = 0, dsum = 0;
  for (int j = ll; j < 128; j += 8) {
    float a = PD[pp*LD_PDPI + j], b = PI[pp*LD_PDPI + j];
    dotp += a*b; nx2 += a*a; ny2 += b*b; dsum += fabsf(a - b);
  }
  dotp = pred8(dotp, RED, pp, ll);
  nx2  = pred8(nx2,  RED, pp, ll);
  ny2  = pred8(ny2,  RED, pp, ll);
  dsum = pred8(dsum, RED, pp, ll);
  float tpart = 0;
  for (int j = ll; j < 128; j += 8)
    tpart += gelu_f(H2[pp*LD_PDPI + j] + P.tb1[j]) * P.tw2[j];
  tpart = pred8(tpart, RED, pp, ll);
  float ts     = P.topo_scale[0] * tanhf(tpart + P.tb2[0]);
  float topo_f = P.scale[0] * ts;
  float nx  = fmaxf(sqrtf(nx2), 1e-8f), ny = fmaxf(sqrtf(ny2), 1e-8f);
  float cosv  = dotp / (nx * ny);
  float fuzzy = expf(-dsum * (1.f/128.f));

  // ---- build full[772], skip = full @ sw, then LN in place ----
  float* Ap = A + pp*LD_A;
  {
    const float* pd = PD + pp*LD_PDPI;
    const float* pi = PI + pp*LD_PDPI;
    for (int j = ll; j < 128; j += 8) {
      float a = pd[j], b = pi[j], df = a - b;
      Ap[j]       = a;
      Ap[128 + j] = b;
      Ap[256 + j] = a * b;
      Ap[384 + j] = fabsf(df);
      Ap[512 + j] = a + b;
      Ap[640 + j] = df * df;
    }
    if (ll == 0) Ap[768] = dotp;
    if (ll == 1) Ap[769] = cosv;
    if (ll == 2) Ap[770] = topo_f;
    if (ll == 3) Ap[771] = fuzzy;
  }
  __syncthreads();
  float s = 0, s2 = 0, sk0 = 0, sk1 = 0;
  for (int j = ll; j < PAIR_DIM; j += 8) {
    float f = Ap[j];
    s += f; s2 += f*f;
    sk0 += f * P.sw[2*j];
    sk1 += f * P.sw[2*j + 1];
  }
  s   = pred8(s,   RED, pp, ll);
  s2  = pred8(s2,  RED, pp, ll);
  sk0 = pred8(sk0, RED, pp, ll);
  sk1 = pred8(sk1, RED, pp, ll);
  if (ll == 0) { SC[pp*4] = sk0; SC[pp*4 + 1] = sk1; }
  {
    float mean = s * (1.f/772.f);
    float varf = s2 * (1.f/772.f) - mean*mean;
    float inv  = rsqrtf(varf + 1e-5f);
    for (int j = ll; j < PAIR_DIM; j += 8)
      Ap[j] = (Ap[j] - mean) * inv * P.ln_g[j] + P.ln_b[j];
  }
  __syncthreads();

  // ---- GEMM1: H1 = gelu(LN(full) @ w1 + b1)   [32x772]@[772x384] ----
  #pragma unroll
  for (int tt = 0; tt < 3; tt++) {
    int n0 = wv*16 + tt*128;
    v8f c0 = {0,0,0,0,0,0,0,0}, c1 = {0,0,0,0,0,0,0,0};
    gemm2m(c0, c1, A, LD_A, P.w1, HID_M, n0, PAIR_DIM, half, l16);
    store2m(H1, LD_H1, n0, c0, c1, P.b1, half, l16, 1);
  }
  __syncthreads();

  // ---- GEMM2: H2 = gelu(H1 @ w2 + b2)   [32x384]@[384x256] ----
  #pragma unroll
  for (int tt = 0; tt < 2; tt++) {
    int n0 = wv*16 + tt*128;
    v8f c0 = {0,0,0,0,0,0,0,0}, c1 = {0,0,0,0,0,0,0,0};
    gemm2m(c0, c1, H1, LD_H1, P.w2, CPT, n0, HID_M, half, l16);
    store2m(H2, LD_H2, n0, c0, c1, P.b2, half, l16, 1);
  }
  __syncthreads();

  // ---- bilinear temps: TMP[k] = pi @ bw[k]^T   (reuses H1 region) ----
  float* TMP = H1;
  #pragma unroll
  for (int k2 = 0; k2 < 2; k2++) {
    int n0 = wv * 16;
    v8f c0 = {0,0,0,0,0,0,0,0}, c1 = {0,0,0,0,0,0,0,0};
    gemm2m_bt(c0, c1, PI, LD_PDPI, P.bw + k2*128*128, 128, n0, 128, half, l16);
    store2m(TMP + k2*(32*LD_PDPI), LD_PDPI, n0, c0, c1, nullptr, half, l16, 0);
  }
  __syncthreads();

  // ---- final combine ----
  float m0 = 0, m1 = 0, b0 = 0, b1v = 0;
  for (int j = ll; j < 256; j += 8) {
    float hv = H2[pp*LD_H2 + j];
    m0 += hv * P.w3[2*j];
    m1 += hv * P.w3[2*j + 1];
  }
  for (int j = ll; j < 128; j += 8) {
    float a = PD[pp*LD_PDPI + j];
    b0  += a * TMP[pp*LD_PDPI + j];
    b1v += a * TMP[32*LD_PDPI + pp*LD_PDPI + j];
  }
  m0  = pred8(m0,  RED, pp, ll);
  m1  = pred8(m1,  RED, pp, ll);
  b0  = pred8(b0,  RED, pp, ll);
  b1v = pred8(b1v, RED, pp, ll);
  if (ll == 0) {
    float main0 = m0 + P.b3[0], main1 = m1 + P.b3[1];
    float bil0  = b0 + P.bb[0], bil1  = b1v + P.bb[1];
    out[2*pg]     = 0.6f*main0 + 0.4f*bil0 + SC[pp*4]     + P.sb[0];
    out[2*pg + 1] = 0.6f*main1 + 0.4f*bil1 + SC[pp*4 + 1] + P.sb[1];
  }
}

// =============================== launch ====================================
// jax tree_flatten order (dict keys sorted, recursive), after the 5 array inputs:
//  5..12  dfuse:  b1,b2,ln_b,ln_g,out_b,out_g,w1,w2
// 13..26  dgate:  nb1,nb2,nln_b,nln_g,nw1,nw2,out_b,out_g,pb1,pb2,pln_b,pln_g,pw1,pw2
// 27..34  difuse, 35..48 digate
// 49..61  pair:   b1,b2,b3,bb,bw,ln_b,ln_g,sb,scale,sw,w1,w2,w3
// 62 tb1, 63 tb2, 64 topo_scale, 65 tw1, 66 tw2
extern "C" void kernel_launch(void* const* d_in, const int* in_sizes, int n_in,
                              void* d_out, int out_size, void* d_ws, size_t ws_size,
                              hipStream_t stream) {
  (void)in_sizes; (void)n_in; (void)out_size; (void)ws_size;
  #define PF(i) ((const float*)d_in[(i)])
  const float* drug_views = PF(0);
  const float* dis_views  = PF(1);
  const float* drug_hgt   = PF(2);
  const float* dis_hgt    = PF(3);
  const int*   sample     = (const int*)d_in[4];

  FuseP dfuse  = { PF(5),  PF(6),  PF(7),  PF(8),  PF(9),  PF(10), PF(11), PF(12) };
  GateP dgate  = { PF(13), PF(14), PF(15), PF(16), PF(17), PF(18), PF(19), PF(20),
                   PF(21), PF(22), PF(23), PF(24), PF(25), PF(26) };
  FuseP difuse = { PF(27), PF(28), PF(29), PF(30), PF(31), PF(32), PF(33), PF(34) };
  GateP digate = { PF(35), PF(36), PF(37), PF(38), PF(39), PF(40), PF(41), PF(42),
                   PF(43), PF(44), PF(45), PF(46), PF(47), PF(48) };
  PairP P      = { PF(49), PF(50), PF(51), PF(52), PF(53), PF(54), PF(55), PF(56),
                   PF(57), PF(58), PF(59), PF(60), PF(61),
                   PF(62), PF(63), PF(64), PF(65), PF(66) };
  #undef PF

  float* drugE = (float*)d_ws;                 // [16384][128]
  float* disE  = drugE + (size_t)N_NODE * DIM; // [16384][128]

  node_kernel<<<N_NODE, 128, 0, stream>>>(drug_views, drug_hgt, dfuse, dgate, drugE);
  node_kernel<<<N_NODE, 128, 0, stream>>>(dis_views,  dis_hgt,  difuse, digate, disE);
  pair_kernel<<<N_PAIR/32, 256, SMEM_BYTES, stream>>>(drugE, disE, sample, P,
                                                      (float*)d_out);
}